// A2CNetwork_61744449847621
// MI455X (gfx1250) — compile-verified
//
#include <hip/hip_runtime.h>

#define GCN_D   64
#define IN_DIM  128
#define RAW_D   (IN_DIM - GCN_D)   // 64
#define OUT_DIM 16
#define BB      16
#define NN      10000
#define EE      160000
#define H1      512
#define H2      1024

typedef __attribute__((ext_vector_type(2))) float v2f;
typedef __attribute__((ext_vector_type(8))) float v8f;

// ---------------------------------------------------------------------------
// Init: deg = 1.0 (self loop), gsum = 0
// ---------------------------------------------------------------------------
__global__ void init_kernel(float* __restrict__ deg, float* __restrict__ gsum) {
    int idx = blockIdx.x * blockDim.x + threadIdx.x;
    if (idx < BB * NN) deg[idx] = 1.0f;
    if (idx < BB * GCN_D) gsum[idx] = 0.0f;
}

// ---------------------------------------------------------------------------
// Degree accumulation: deg[b, dst] += 1 per edge
// ---------------------------------------------------------------------------
__global__ void deg_acc_kernel(const int* __restrict__ ei, float* __restrict__ deg) {
    const int b = blockIdx.y;
    const int e = blockIdx.x * blockDim.x + threadIdx.x;
    if (e < EE) {
        int d = ei[(size_t)b * 2 * EE + EE + e];
        atomicAdd(&deg[(size_t)b * NN + d], 1.0f);
    }
}

// dis = rsqrt(deg)
__global__ void dis_kernel(const float* __restrict__ deg, float* __restrict__ dis) {
    int idx = blockIdx.x * blockDim.x + threadIdx.x;
    if (idx < BB * NN) dis[idx] = rsqrtf(deg[idx]);
}

// ---------------------------------------------------------------------------
// xw[b] = gcn_x[b] @ W  via V_WMMA_F32_16X16X4_F32.
// W (16 KB) is staged TRANSPOSED in LDS once per workgroup so each B
// fragment is one ds_load_b64 (two K-adjacent values contiguous).
// One wave computes a 16-row x 64-col strip (4 accumulators), K-loop step 4.
// ---------------------------------------------------------------------------
__global__ __launch_bounds__(128) void xw_wmma_kernel(const float* __restrict__ gcn_x,
                                                      const float* __restrict__ W,
                                                      float* __restrict__ xw) {
    __shared__ float shWt[GCN_D * GCN_D];   // shWt[col*64 + k] = W[k*64 + col]

    const int b    = blockIdx.y;
    const int wave = threadIdx.x >> 5;
    const int lane = threadIdx.x & 31;

    // Stage W transposed: 4096 elements / 128 threads = 32 each.
    for (int i = threadIdx.x; i < GCN_D * GCN_D; i += 128) {
        const int r = i >> 6, c = i & 63;          // W row-major [k=r][col=c]
        shWt[c * GCN_D + r] = W[i];
    }
    __syncthreads();

    const int mtile = blockIdx.x * 4 + wave;       // wave-uniform
    if (mtile * 16 < NN) {
        const int m0 = mtile * 16;
        const float* A = gcn_x + ((size_t)b * NN + m0) * GCN_D;
        float*       D = xw    + ((size_t)b * NN + m0) * GCN_D;

        const int arow = lane & 15;
        const int koff = (lane >> 4) * 2;

        v8f zero = {0.f, 0.f, 0.f, 0.f, 0.f, 0.f, 0.f, 0.f};
        v8f acc0 = zero, acc1 = zero, acc2 = zero, acc3 = zero;

        #pragma unroll
        for (int k = 0; k < GCN_D; k += 4) {
            const v2f a = *(const v2f*)&A[arow * GCN_D + k + koff];

            const int colb = lane & 15;
            const v2f b0 = *(const v2f*)&shWt[( 0 + colb) * GCN_D + k + koff];
            const v2f b1 = *(const v2f*)&shWt[(16 + colb) * GCN_D + k + koff];
            const v2f b2 = *(const v2f*)&shWt[(32 + colb) * GCN_D + k + koff];
            const v2f b3 = *(const v2f*)&shWt[(48 + colb) * GCN_D + k + koff];

            acc0 = __builtin_amdgcn_wmma_f32_16x16x4_f32(false, a, false, b0, (short)0, acc0, false, false);
            acc1 = __builtin_amdgcn_wmma_f32_16x16x4_f32(false, a, false, b1, (short)0, acc1, false, false);
            acc2 = __builtin_amdgcn_wmma_f32_16x16x4_f32(false, a, false, b2, (short)0, acc2, false, false);
            acc3 = __builtin_amdgcn_wmma_f32_16x16x4_f32(false, a, false, b3, (short)0, acc3, false, false);
        }

        const int rbase = (lane >> 4) * 8;
        const int cbase = lane & 15;
        #pragma unroll
        for (int i = 0; i < 8; ++i) {
            D[(rbase + i) * GCN_D +  0 + cbase] = acc0[i];
            D[(rbase + i) * GCN_D + 16 + cbase] = acc1[i];
            D[(rbase + i) * GCN_D + 32 + cbase] = acc2[i];
            D[(rbase + i) * GCN_D + 48 + cbase] = acc3[i];
        }
    }
}

// ---------------------------------------------------------------------------
// Edge message + pool fused: gsum[b,:] += sum_e xw[b,src]*dis[src]*dis[dst]
// Wave processes 32 edges per chunk: each lane loads one edge's (src,dst)
// coalesced and computes its norm; inner loop shfl-broadcasts (src,norm)
// and issues independent 256B/wave gathers (lane holds dims 2l,2l+1).
// Block-level LDS reduction (ds_add_f32) before global atomics.
// EE % 32 == 0 so chunking is exact.
// ---------------------------------------------------------------------------
__global__ __launch_bounds__(256) void edge_sum_kernel(const float* __restrict__ xw,
                                                       const int* __restrict__ ei,
                                                       const float* __restrict__ dis,
                                                       float* __restrict__ gsum) {
    __shared__ float red[GCN_D];

    const int b    = blockIdx.y;
    const int lane = threadIdx.x & 31;
    const int wid  = blockIdx.x * (blockDim.x >> 5) + (threadIdx.x >> 5);
    const int nw   = gridDim.x * (blockDim.x >> 5);

    const int*   src_arr = ei + (size_t)b * 2 * EE;
    const int*   dst_arr = src_arr + EE;
    const float* disb    = dis + (size_t)b * NN;
    const float* xwb     = xw + (size_t)b * NN * GCN_D;

    if (threadIdx.x < GCN_D) red[threadIdx.x] = 0.f;
    __syncthreads();

    float a0 = 0.f, a1 = 0.f;
    const int nchunk = EE / 32;
    for (int c = wid; c < nchunk; c += nw) {
        const int e = c * 32 + lane;
        __builtin_prefetch(&src_arr[e + nw * 32], 0, 0);   // global_prefetch_b8
        __builtin_prefetch(&dst_arr[e + nw * 32], 0, 0);
        const int   s   = src_arr[e];
        const int   d   = dst_arr[e];
        const float nrm = disb[s] * disb[d];

        #pragma unroll 8
        for (int j = 0; j < 32; ++j) {
            const int   sj = __shfl(s, j, 32);
            const float nj = __shfl(nrm, j, 32);
            const float2 v = *(((const float2*)(xwb + (size_t)sj * GCN_D)) + lane);
            a0 += v.x * nj;
            a1 += v.y * nj;
        }
    }

    atomicAdd(&red[2 * lane    ], a0);     // ds_add_f32
    atomicAdd(&red[2 * lane + 1], a1);
    __syncthreads();
    if (threadIdx.x < GCN_D) atomicAdd(&gsum[b * GCN_D + threadIdx.x], red[threadIdx.x]);
}

// Self-loop term: gsum[b,:] += sum_n xw[b,n] / deg[b,n]
__global__ __launch_bounds__(256) void self_sum_kernel(const float* __restrict__ xw,
                                                       const float* __restrict__ dis,
                                                       float* __restrict__ gsum) {
    __shared__ float red[GCN_D];

    const int b    = blockIdx.y;
    const int lane = threadIdx.x & 31;
    const int wid  = blockIdx.x * (blockDim.x >> 5) + (threadIdx.x >> 5);
    const int nw   = gridDim.x * (blockDim.x >> 5);

    const float* disb = dis + (size_t)b * NN;
    const float* xwb  = xw + (size_t)b * NN * GCN_D;

    if (threadIdx.x < GCN_D) red[threadIdx.x] = 0.f;
    __syncthreads();

    float a0 = 0.f, a1 = 0.f;
    for (int n = wid; n < NN; n += nw) {
        const float di  = disb[n];
        const float nrm = di * di;           // = 1/deg
        const float2 v = *(((const float2*)(xwb + (size_t)n * GCN_D)) + lane);
        a0 += v.x * nrm;
        a1 += v.y * nrm;
    }

    atomicAdd(&red[2 * lane    ], a0);
    atomicAdd(&red[2 * lane + 1], a1);
    __syncthreads();
    if (threadIdx.x < GCN_D) atomicAdd(&gsum[b * GCN_D + threadIdx.x], red[threadIdx.x]);
}

// h = concat(gsum/N + gcn_b, x)  -> [B, IN_DIM]
__global__ void hbuild_kernel(const float* __restrict__ gsum,
                              const float* __restrict__ gcn_b,
                              const float* __restrict__ x,
                              float* __restrict__ h) {
    int idx = blockIdx.x * blockDim.x + threadIdx.x;
    if (idx < BB * IN_DIM) {
        const int bb = idx / IN_DIM;
        const int j  = idx % IN_DIM;
        float v;
        if (j < GCN_D) v = gsum[bb * GCN_D + j] * (1.0f / NN) + gcn_b[j];
        else           v = x[bb * RAW_D + (j - GCN_D)];
        h[idx] = v;
    }
}

// ---------------------------------------------------------------------------
// MLP layer: out[16,Nout] = act(A[16,K] @ Bw[K,Nld] + bias), one wave per
// 16x16 output tile, f32 WMMA K-loop. Nld is the leading dim of Bw; columns
// >= Nout are zero-padded (critic head, Nout=1).
// ---------------------------------------------------------------------------
__global__ __launch_bounds__(32) void mlp_wmma_kernel(const float* __restrict__ A,
                                                      const float* __restrict__ Bw,
                                                      const float* __restrict__ bias,
                                                      float* __restrict__ out,
                                                      int K, int Nld, int Nout, int leaky) {
    const int lane = threadIdx.x & 31;
    const int n0   = blockIdx.x * 16;
    const int arow = lane & 15;
    const int koff = (lane >> 4) * 2;
    const int col  = n0 + (lane & 15);
    const bool colv = (col < Nout);

    v8f acc = {0.f, 0.f, 0.f, 0.f, 0.f, 0.f, 0.f, 0.f};

    for (int k = 0; k < K; k += 4) {
        const v2f a = *(const v2f*)&A[arow * K + k + koff];
        v2f bv;
        bv.x = colv ? Bw[(k + koff)     * Nld + col] : 0.f;
        bv.y = colv ? Bw[(k + koff + 1) * Nld + col] : 0.f;
        acc = __builtin_amdgcn_wmma_f32_16x16x4_f32(
            false, a, false, bv, (short)0, acc, false, false);
    }

    if (colv) {
        const int rbase = (lane >> 4) * 8;
        const float bb = bias[col];
        #pragma unroll
        for (int i = 0; i < 8; ++i) {
            float v = acc[i] + bb;
            if (leaky && v < 0.f) v *= 0.01f;
            out[(rbase + i) * Nout + col] = v;
        }
    }
}

// ---------------------------------------------------------------------------
extern "C" void kernel_launch(void* const* d_in, const int* in_sizes, int n_in,
                              void* d_out, int out_size, void* d_ws, size_t ws_size,
                              hipStream_t stream) {
    const float* x      = (const float*)d_in[0];
    const float* gcn_x  = (const float*)d_in[1];
    const int*   ei     = (const int*)  d_in[2];
    const float* gcn_w  = (const float*)d_in[3];
    const float* gcn_b  = (const float*)d_in[4];
    const float* aw1    = (const float*)d_in[5];
    const float* ab1    = (const float*)d_in[6];
    const float* aw2    = (const float*)d_in[7];
    const float* ab2    = (const float*)d_in[8];
    const float* aw3    = (const float*)d_in[9];
    const float* ab3    = (const float*)d_in[10];
    const float* cw1    = (const float*)d_in[11];
    const float* cb1    = (const float*)d_in[12];
    const float* cw2    = (const float*)d_in[13];
    const float* cb2    = (const float*)d_in[14];
    const float* cw3    = (const float*)d_in[15];
    const float* cb3    = (const float*)d_in[16];

    float* ws   = (float*)d_ws;
    float* xw   = ws;                                   // B*N*64
    float* deg  = xw   + (size_t)BB * NN * GCN_D;       // B*N
    float* dis  = deg  + (size_t)BB * NN;               // B*N
    float* gsum = dis  + (size_t)BB * NN;               // B*64
    float* h    = gsum + (size_t)BB * GCN_D;            // B*128
    float* a1b  = h    + (size_t)BB * IN_DIM;           // B*H1
    float* a2b  = a1b  + (size_t)BB * H1;               // B*H2
    float* c1b  = a2b  + (size_t)BB * H2;               // B*H1
    float* c2b  = c1b  + (size_t)BB * H1;               // B*H2

    float* outA = (float*)d_out;          // [16,16]
    float* outC = outA + BB * OUT_DIM;    // [16,1]

    // 1) init deg=1, gsum=0
    init_kernel<<<(BB * NN + 255) / 256, 256, 0, stream>>>(deg, gsum);
    // 2) degree counts
    deg_acc_kernel<<<dim3((EE + 255) / 256, BB), 256, 0, stream>>>(ei, deg);
    // 3) dis = rsqrt(deg)
    dis_kernel<<<(BB * NN + 255) / 256, 256, 0, stream>>>(deg, dis);
    // 4) xw = gcn_x @ W  (WMMA f32, W staged transposed in LDS)
    xw_wmma_kernel<<<dim3((NN / 16 + 3) / 4, BB), 128, 0, stream>>>(gcn_x, gcn_w, xw);
    // 5) fused edge message + mean-pool partial sums
    edge_sum_kernel<<<dim3(64, BB), 256, 0, stream>>>(xw, ei, dis, gsum);
    // 6) self-loop term
    self_sum_kernel<<<dim3(16, BB), 256, 0, stream>>>(xw, dis, gsum);
    // 7) h = [gsum/N + b, x]
    hbuild_kernel<<<(BB * IN_DIM + 255) / 256, 256, 0, stream>>>(gsum, gcn_b, x, h);
    // 8) actor MLP
    mlp_wmma_kernel<<<H1 / 16, 32, 0, stream>>>(h,   aw1, ab1, a1b,  IN_DIM, H1, H1, 1);
    mlp_wmma_kernel<<<H2 / 16, 32, 0, stream>>>(a1b, aw2, ab2, a2b,  H1,     H2, H2, 1);
    mlp_wmma_kernel<<<1,       32, 0, stream>>>(a2b, aw3, ab3, outA, H2, OUT_DIM, OUT_DIM, 0);
    // 9) critic MLP
    mlp_wmma_kernel<<<H1 / 16, 32, 0, stream>>>(h,   cw1, cb1, c1b,  IN_DIM, H1, H1, 1);
    mlp_wmma_kernel<<<H2 / 16, 32, 0, stream>>>(c1b, cw2, cb2, c2b,  H1,     H2, H2, 1);
    mlp_wmma_kernel<<<1,       32, 0, stream>>>(c2b, cw3, cb3, outC, H2, 1, 1, 0);
}